// StreamingSTFT_56951266345124
// MI455X (gfx1250) — compile-verified
//
#include <hip/hip_runtime.h>

typedef float v4f __attribute__((ext_vector_type(4)));

#define HOP   64
#define DELAY 192   // 3 * HOP: latency of the analysis->OLA pipeline

// y[i*64+k] = chunk[(i-3)*64+k] * ( ana[k+64]*syn[k+64] + ana[k+128]*syn[k+128] )
// (rfft -> irfft is the identity; OLA telescopes to a delayed diagonal scale)
__global__ __launch_bounds__(256) void stft_ola_diag_kernel(
    const float* __restrict__ chunk,
    const float* __restrict__ ana,
    const float* __restrict__ syn,
    float* __restrict__ out,
    int total) {
  __shared__ float tile[256 * 4];

  const int tid = threadIdx.x;
  const int n   = (blockIdx.x * 256 + tid) * 4;   // flat output element index
  if (n >= total) return;

  const int s  = n - DELAY;                        // delayed source index
  const int sc = s < 0 ? 0 : s;                    // clamp: keeps EXEC uniform; masked below

  // ---- CDNA5 async copy: global -> LDS, 16 B per lane, tracked by ASYNCcnt
  unsigned lds_addr = (unsigned)(unsigned long long)(&tile[tid * 4]);
  const float* gp = chunk + sc;
  asm volatile("global_load_async_to_lds_b128 %0, %1, off"
               :: "v"(lds_addr), "v"(gp)
               : "memory");

  // ---- combined OLA window (j=1 + j=2 overlap terms), hidden under async latency
  const int k0 = n & (HOP - 1);                    // multiple of 4, so each v4 stays in-frame
  v4f a1 = *(const v4f*)(ana + k0 + 64);
  v4f s1 = *(const v4f*)(syn + k0 + 64);
  v4f a2 = *(const v4f*)(ana + k0 + 128);
  v4f s2 = *(const v4f*)(syn + k0 + 128);
  v4f w  = a1 * s1 + a2 * s2;

  // ---- wait on this wave's async transfers, read back own 16 B from LDS
  asm volatile("s_wait_asynccnt 0" ::: "memory");
  v4f x = *(const v4f*)(&tile[tid * 4]);

  v4f zero = {0.f, 0.f, 0.f, 0.f};
  v4f y = (s >= 0) ? x * w : zero;

  // write-once stream: non-temporal b128 store, don't pollute L2
  __builtin_nontemporal_store(y, (v4f*)(out + n));
}

extern "C" void kernel_launch(void* const* d_in, const int* in_sizes, int n_in,
                              void* d_out, int out_size, void* d_ws, size_t ws_size,
                              hipStream_t stream) {
  const float* chunk = (const float*)d_in[0];   // [131072, 64] f32
  const float* ana   = (const float*)d_in[1];   // [256] f32
  const float* syn   = (const float*)d_in[2];   // [256] f32
  float* out = (float*)d_out;                   // [131072, 64] f32

  const int total           = out_size;         // 8388608 elements
  const int threads         = 256;              // 8 wave32 per block
  const int elems_per_block = threads * 4;      // 16 B per lane
  const int blocks          = (total + elems_per_block - 1) / elems_per_block;

  stft_ola_diag_kernel<<<blocks, threads, 0, stream>>>(chunk, ana, syn, out, total);
}